// Net_81612968558774
// MI455X (gfx1250) — compile-verified
//
#include <hip/hip_runtime.h>
#include <hip/hip_bf16.h>
#include <float.h>

// ---------------- problem constants (match reference) ----------------
#define BATCH      4
#define NPG        8192
#define NPTS       32768          // BATCH*NPG
#define EDGES      524288         // NPTS*16
#define KSPL       5
#define KD         125            // KSPL^3
#define SLOTS      64
#define NCLS       101

typedef __attribute__((ext_vector_type(2))) float v2f;
typedef __attribute__((ext_vector_type(8))) float v8f;

// ---------------- helpers ----------------
static __device__ __forceinline__ unsigned enc_ord(float f) {
    unsigned u = __float_as_uint(f);
    return (u & 0x80000000u) ? ~u : (u | 0x80000000u);
}
static __device__ __forceinline__ float dec_ord(unsigned u) {
    return (u & 0x80000000u) ? __uint_as_float(u & 0x7fffffffu)
                             : __uint_as_float(~u);
}
#define ENC_NEG_FLT_MAX 0x00800000u   // enc_ord(-FLT_MAX)

__global__ void fill_f32(float* p, float v, long long n) {
    long long i = (long long)blockIdx.x * blockDim.x + threadIdx.x;
    if (i < n) p[i] = v;
}
__global__ void fill_u32(unsigned* p, unsigned v, long long n) {
    long long i = (long long)blockIdx.x * blockDim.x + threadIdx.x;
    if (i < n) p[i] = v;
}

// ---------------- edge attr (cartesian pseudo-coords) ----------------
__global__ void edge_absmax(const float* __restrict__ pos, const int* __restrict__ s,
                            const int* __restrict__ d, const float* __restrict__ em,
                            unsigned* mbits, int nE) {
    __shared__ float red[256];
    int e = blockIdx.x * blockDim.x + threadIdx.x;
    float mx = 0.f;
    if (e < nE) {
        float w = em ? em[e] : 1.f;
        int si = s[e], di = d[e];
        #pragma unroll
        for (int k = 0; k < 3; ++k)
            mx = fmaxf(mx, fabsf(pos[di*3+k] - pos[si*3+k]) * w);
    }
    red[threadIdx.x] = mx;
    __syncthreads();
    for (int o = 128; o > 0; o >>= 1) {
        if ((int)threadIdx.x < o) red[threadIdx.x] = fmaxf(red[threadIdx.x], red[threadIdx.x + o]);
        __syncthreads();
    }
    if (threadIdx.x == 0) atomicMax(mbits, __float_as_uint(red[0])); // values >= 0
}

__global__ void edge_attr(const float* __restrict__ pos, const int* __restrict__ s,
                          const int* __restrict__ d, const unsigned* mbits,
                          float* __restrict__ attr, int nE) {
    int e = blockIdx.x * blockDim.x + threadIdx.x;
    if (e >= nE) return;
    float m = __uint_as_float(*mbits);
    float inv = 1.f / (2.f * m + 1e-12f);
    int si = s[e], di = d[e];
    #pragma unroll
    for (int k = 0; k < 3; ++k)
        attr[(size_t)e*3+k] = (pos[di*3+k] - pos[si*3+k]) * inv + 0.5f;
}

// ---------------- spline basis scatter:  A[(dst*KD+idx)*cin + c] += b*x[src,c] ----
__global__ void spline_scatter(const float* __restrict__ x, const float* __restrict__ attr,
                               const int* __restrict__ s, const int* __restrict__ d,
                               const float* __restrict__ em, float* __restrict__ A,
                               float* __restrict__ deg, int nE, int cin) {
    long long gid = (long long)blockIdx.x * blockDim.x + threadIdx.x;
    if (gid >= (long long)nE * cin) return;
    int e = (int)(gid / cin);
    int c = (int)(gid % cin);
    float w = em ? em[e] : 1.f;
    if (w <= 0.f) return;
    int di = d[e];
    if (c == 0) atomicAdd(&deg[di], w);
    float xv = x[(size_t)s[e]*cin + c];
    float fr[3]; int lo[3];
    #pragma unroll
    for (int k = 0; k < 3; ++k) {
        float v  = fminf(fmaxf(attr[(size_t)e*3+k], 0.f), 1.f) * (float)(KSPL-1);
        float fl = fminf(floorf(v), (float)(KSPL-2));
        lo[k] = (int)fl; fr[k] = v - fl;
    }
    size_t base = (size_t)di * KD * cin + c;
    #pragma unroll
    for (int bits = 0; bits < 8; ++bits) {
        float b = w; int idx = 0, stride = 1;
        #pragma unroll
        for (int k = 0; k < 3; ++k) {
            int sb = (bits >> k) & 1;
            b *= sb ? fr[k] : (1.f - fr[k]);
            idx += (lo[k] + sb) * stride; stride *= KSPL;
        }
        atomicAdd(&A[base + (size_t)idx * cin], b * xv);
    }
}

// ---------------- fp32 WMMA GEMM (LDS tiled):  C(M,N) = A(M,K) @ B(K,N) ------
// Block: 256 thr / 8 waves. Block tile 128(M) x 64(N), K-chunk 32.
// A kept row-major in LDS (stride 34), B stored TRANSPOSED (BsT[col*34+k]) so
// each WMMA fragment is one aligned ds_load_b64 (ka always even, stride even).
// Inner loop: 8 k-steps x 4 tiles = 32 v_wmma per stage, no register repacking.
#define TM 128
#define TN 64
#define TK 32
#define APAD 34      // even stride: (row*34+ka) even -> b64-aligned fragments
#define BPAD 34      // BsT[col*34+k], same property

__global__ __launch_bounds__(256) void wmma_gemm_f32(
        const float* __restrict__ Am, const float* __restrict__ Bm,
        float* __restrict__ Cm, int M, int Kdim, int Ndim, int kchunk) {
    __shared__ float As[TM * APAD];       // 17408 B
    __shared__ float BsT[TN * BPAD];      //  8704 B
    const int tid  = threadIdx.x;
    const int lane = tid & 31;
    const int wave = tid >> 5;
    const int half = lane >> 4;
    const int l16  = lane & 15;
    const int mbase = blockIdx.y * TM;
    const int n0    = blockIdx.x * TN;
    const int kbeg  = blockIdx.z * kchunk;
    const int kend  = (kbeg + kchunk < Kdim) ? (kbeg + kchunk) : Kdim;
    const int m0    = mbase + wave * 16;

    v8f acc[4] = {v8f{}, v8f{}, v8f{}, v8f{}};

    for (int k0 = kbeg; k0 < kend; k0 += TK) {
        // ---- stage A tile (128x32): coalesced along K, clamp+select (no branches) ----
        #pragma unroll
        for (int j = 0; j < (TM * TK) / 256; ++j) {
            int q   = tid + 256 * j;
            int row = q >> 5, kk = q & 31;
            int gm  = mbase + row;
            int gk  = k0 + kk;
            bool ok = (gm < M) & (gk < kend);
            int gm_c = (gm < M) ? gm : (M - 1);
            int gk_c = (gk < kend) ? gk : (kend - 1);
            float v = Am[(size_t)gm_c * Kdim + gk_c];
            As[row * APAD + kk] = ok ? v : 0.f;
        }
        // ---- stage B tile (32x64) transposed into LDS: coalesced along N ----
        #pragma unroll
        for (int j = 0; j < (TK * TN) / 256; ++j) {
            int q   = tid + 256 * j;
            int col = q & 63, row = q >> 6;
            int gk  = k0 + row;
            int gc  = n0 + col;
            bool ok = (gk < kend) & (gc < Ndim);
            int gk_c = (gk < kend) ? gk : (kend - 1);
            int gc_c = (gc < Ndim) ? gc : (Ndim - 1);
            float v = Bm[(size_t)gk_c * Ndim + gc_c];
            BsT[col * BPAD + row] = ok ? v : 0.f;
        }
        __syncthreads();

        // ---- 8 k-steps x 4 N-tiles of v_wmma_f32_16x16x4_f32 ----
        const float* arow = &As[(wave * 16 + l16) * APAD];
        #pragma unroll
        for (int kk = 0; kk < TK; kk += 4) {
            const int ka = kk + 2 * half;          // even: aligned b64 fragment
            v2f a = *(const v2f*)(arow + ka);
            #pragma unroll
            for (int t = 0; t < 4; ++t) {
                v2f b = *(const v2f*)(&BsT[(t * 16 + l16) * BPAD] + ka);
                acc[t] = __builtin_amdgcn_wmma_f32_16x16x4_f32(
                             false, a, false, b, (short)0, acc[t], false, false);
            }
        }
        __syncthreads();
    }

    const bool atom = gridDim.z > 1;
    #pragma unroll
    for (int t = 0; t < 4; ++t) {
        const int col = n0 + t * 16 + l16;
        if (col < Ndim) {
            #pragma unroll
            for (int r = 0; r < 8; ++r) {
                const int row = m0 + half * 8 + r;
                if (row < M) {
                    if (atom) atomicAdd(&Cm[(size_t)row * Ndim + col], acc[t][r]);
                    else      Cm[(size_t)row * Ndim + col] = acc[t][r];
                }
            }
        }
    }
}

// ---------------- conv epilogue: /deg + x@root + bias, ELU --------------
__global__ void post_conv(const float* __restrict__ conv, const float* __restrict__ xprev,
                          const float* __restrict__ root, const float* __restrict__ bias,
                          const float* __restrict__ deg, float* __restrict__ tout,
                          int n, int cin, int cout) {
    long long gid = (long long)blockIdx.x * blockDim.x + threadIdx.x;
    if (gid >= (long long)n * cout) return;
    int i  = (int)(gid / cout);
    int co = (int)(gid % cout);
    float dd = fmaxf(deg[i], 1.f);
    float acc = conv[gid] / dd + bias[co];
    for (int c = 0; c < cin; ++c)
        acc += xprev[(size_t)i * cin + c] * root[(size_t)c * cout + co];
    tout[gid] = acc > 0.f ? acc : expm1f(acc);   // ELU
}

// ---------------- batch-norm (masked) ----------------
__global__ void bn_stats(const float* __restrict__ t, const float* __restrict__ nmask,
                         float* __restrict__ sum, float* __restrict__ sumsq,
                         float* __restrict__ cnt, int n, int cout) {
    __shared__ float s0[256], s1[256], s2[256];
    int co = blockIdx.x;
    float a = 0.f, b = 0.f, c = 0.f;
    for (int i = threadIdx.x; i < n; i += blockDim.x) {
        float m = nmask ? nmask[i] : 1.f;
        float v = t[(size_t)i * cout + co];
        a += v * m; b += v * v * m; c += m;
    }
    s0[threadIdx.x] = a; s1[threadIdx.x] = b; s2[threadIdx.x] = c;
    __syncthreads();
    for (int o = 128; o > 0; o >>= 1) {
        if ((int)threadIdx.x < o) {
            s0[threadIdx.x] += s0[threadIdx.x + o];
            s1[threadIdx.x] += s1[threadIdx.x + o];
            s2[threadIdx.x] += s2[threadIdx.x + o];
        }
        __syncthreads();
    }
    if (threadIdx.x == 0) { sum[co] = s0[0]; sumsq[co] = s1[0]; if (co == 0) cnt[0] = s2[0]; }
}

__global__ void bn_apply(const float* __restrict__ t, const float* __restrict__ nmask,
                         const float* __restrict__ sum, const float* __restrict__ sumsq,
                         const float* __restrict__ cnt, const float* __restrict__ g,
                         const float* __restrict__ be, float* __restrict__ xout,
                         int n, int cout) {
    long long gid = (long long)blockIdx.x * blockDim.x + threadIdx.x;
    if (gid >= (long long)n * cout) return;
    int i  = (int)(gid / cout);
    int co = (int)(gid % cout);
    float C  = fmaxf(cnt[0], 1.f);
    float mu = sum[co] / C;
    float var = fmaxf(sumsq[co] / C - mu * mu, 0.f);
    float m = nmask ? nmask[i] : 1.f;
    xout[gid] = (g[co] * (t[gid] - mu) * rsqrtf(var + 1e-5f) + be[co]) * m;
}

// ---------------- voxel pooling ----------------
__global__ void pos_min(const float* __restrict__ pos, const float* __restrict__ nmask,
                        unsigned* startbits, int n) {
    int i = blockIdx.x * blockDim.x + threadIdx.x;
    if (i >= n) return;
    float m = nmask ? nmask[i] : 1.f;
    if (m <= 0.f) return;
    #pragma unroll
    for (int k = 0; k < 3; ++k)
        atomicMin(&startbits[k], __float_as_uint(pos[(size_t)i*3+k]));  // pos >= 0
}

__global__ void pool_node(const float* __restrict__ pos, const float* __restrict__ nmask,
                          const unsigned* __restrict__ startbits, int* __restrict__ cl,
                          float* __restrict__ cnt, float* __restrict__ psum,
                          int n, int batch_div, int slot_mult, int G, float size) {
    int i = blockIdx.x * blockDim.x + threadIdx.x;
    if (i >= n) return;
    float m = nmask ? nmask[i] : 1.f;
    int idx[3];
    #pragma unroll
    for (int k = 0; k < 3; ++k) {
        float st = __uint_as_float(startbits[k]);
        float v = floorf((pos[(size_t)i*3+k] - st) / size);
        v = fminf(fmaxf(v, 0.f), (float)(G - 1));
        idx[k] = (int)v;
    }
    int vid = idx[0] + G * idx[1] + G * G * idx[2];
    int c = (i / batch_div) * slot_mult + vid;
    cl[i] = c;
    if (m > 0.f) {
        atomicAdd(&cnt[c], m);
        #pragma unroll
        for (int k = 0; k < 3; ++k)
            atomicAdd(&psum[(size_t)c*3+k], pos[(size_t)i*3+k] * m);
    }
}

__global__ void pool_xmax(const float* __restrict__ x, const float* __restrict__ nmask,
                          const int* __restrict__ cl, unsigned* __restrict__ nxenc,
                          int n, int cout) {
    long long gid = (long long)blockIdx.x * blockDim.x + threadIdx.x;
    if (gid >= (long long)n * cout) return;
    int i  = (int)(gid / cout);
    int co = (int)(gid % cout);
    float m = nmask ? nmask[i] : 1.f;
    if (m <= 0.f) return;
    atomicMax(&nxenc[(size_t)cl[i] * cout + co], enc_ord(x[gid]));
}

__global__ void pool_fin_x(const unsigned* __restrict__ nxenc, const float* __restrict__ cnt,
                           float* __restrict__ xout, int Nn, int cout) {
    long long gid = (long long)blockIdx.x * blockDim.x + threadIdx.x;
    if (gid >= (long long)Nn * cout) return;
    int c = (int)(gid / cout);
    xout[gid] = (cnt[c] > 0.f) ? dec_ord(nxenc[gid]) : 0.f;
}

__global__ void pool_fin_node(const float* __restrict__ cnt, const float* __restrict__ psum,
                              float* __restrict__ posout, float* __restrict__ nmout, int Nn) {
    int c = blockIdx.x * blockDim.x + threadIdx.x;
    if (c >= Nn) return;
    float cc = cnt[c];
    nmout[c] = cc > 0.f ? 1.f : 0.f;
    float inv = 1.f / fmaxf(cc, 1.f);
    #pragma unroll
    for (int k = 0; k < 3; ++k)
        posout[(size_t)c*3+k] = psum[(size_t)c*3+k] * inv;
}

__global__ void edge_remap(const int* __restrict__ s_old, const int* __restrict__ d_old,
                           const float* __restrict__ em_old, const int* __restrict__ cl,
                           int* __restrict__ s_new, int* __restrict__ d_new,
                           float* __restrict__ em_new, unsigned* __restrict__ bitmap,
                           int nE, int Nn) {
    int e = blockIdx.x * blockDim.x + threadIdx.x;
    if (e >= nE) return;
    int s2 = cl[s_old[e]], d2 = cl[d_old[e]];
    s_new[e] = s2; d_new[e] = d2;
    float v = 0.f;
    float w = em_old ? em_old[e] : 1.f;
    if (w > 0.f && s2 != d2) {
        long long key = (long long)s2 * Nn + d2;
        unsigned bit = 1u << (key & 31);
        unsigned old = atomicOr(&bitmap[key >> 5], bit);
        if (!(old & bit)) v = 1.f;              // exactly one edge per distinct pair
    }
    em_new[e] = v;
}

// ---------------- FC head ----------------
__global__ void fc_elu(const float* __restrict__ c, const float* __restrict__ b,
                       float* __restrict__ h, int M, int Nn) {
    int gid = blockIdx.x * blockDim.x + threadIdx.x;
    if (gid >= M * Nn) return;
    float v = c[gid] + b[gid % Nn];
    h[gid] = v > 0.f ? v : expm1f(v);
}

__global__ void head_logsoftmax(const float* __restrict__ l2c, const float* __restrict__ fb2,
                                float* __restrict__ out) {
    int wave = threadIdx.x >> 5, lane = threadIdx.x & 31;
    if (wave >= BATCH) return;
    float mx = -FLT_MAX;
    for (int j = lane; j < NCLS; j += 32) mx = fmaxf(mx, l2c[wave * NCLS + j] + fb2[j]);
    for (int o = 16; o > 0; o >>= 1) mx = fmaxf(mx, __shfl_xor(mx, o));
    float s = 0.f;
    for (int j = lane; j < NCLS; j += 32) s += expf(l2c[wave * NCLS + j] + fb2[j] - mx);
    for (int o = 16; o > 0; o >>= 1) s += __shfl_xor(s, o);
    float ls = logf(s);
    for (int j = lane; j < NCLS; j += 32)
        out[wave * NCLS + j] = l2c[wave * NCLS + j] + fb2[j] - mx - ls;
}

// =================== host orchestration ===================
static inline unsigned gdiv(long long n, int b) { return (unsigned)((n + b - 1) / b); }

extern "C" void kernel_launch(void* const* d_in, const int* in_sizes, int n_in,
                              void* d_out, int out_size, void* d_ws, size_t ws_size,
                              hipStream_t stream) {
    const float* x0   = (const float*)d_in[0];
    const float* pos0 = (const float*)d_in[1];
    const int*   src0 = (const int*)d_in[2];
    const int*   dst0 = (const int*)d_in[3];
    const float *W[4], *R[4], *Bi[4], *Gm[4], *Be[4];
    for (int l = 0; l < 4; ++l) {
        W[l]  = (const float*)d_in[5 + 5*l + 0];
        R[l]  = (const float*)d_in[5 + 5*l + 1];
        Bi[l] = (const float*)d_in[5 + 5*l + 2];
        Gm[l] = (const float*)d_in[5 + 5*l + 3];
        Be[l] = (const float*)d_in[5 + 5*l + 4];
    }
    const float* fw1 = (const float*)d_in[25];
    const float* fb1 = (const float*)d_in[26];
    const float* fw2 = (const float*)d_in[27];
    const float* fb2 = (const float*)d_in[28];

    // ---- workspace bump allocator ----
    size_t off = 0;
    auto alloc = [&](size_t bytes) -> char* {
        char* p = (char*)d_ws + off;
        off += (bytes + 255) & ~(size_t)255;
        return p;
    };
    const int nlist[4]  = {NPTS, 1372, 500, 108};
    const int cinl[4]   = {1, 64, 128, 256};
    const int coutl[4]  = {64, 128, 256, 512};
    const int GRD[3]    = {7, 5, 3};
    const float VOX[3]  = {20.f, 30.f, 50.f};
    const int cpgl[3]   = {343, 125, 27};
    const int Nnl[3]    = {1372, 500, 108};

    const size_t MAXNC = (size_t)NPTS * 64;          // biggest (n*cout)
    float*    attr   = (float*)alloc((size_t)EDGES * 3 * 4);
    float*    Abuf   = (float*)alloc((size_t)1372 * KD * 64 * 4);   // largest A: 10.976M f
    float*    conv   = (float*)alloc(MAXNC * 4);
    float*    tbuf   = (float*)alloc(MAXNC * 4);
    float*    xA     = (float*)alloc(MAXNC * 4);
    float*    xB     = (float*)alloc(MAXNC * 4);
    float*    deg    = (float*)alloc((size_t)NPTS * 4);
    float*    bnsum  = (float*)alloc(512 * 4);
    float*    bnsq   = (float*)alloc(512 * 4);
    float*    bncnt  = (float*)alloc(4);
    unsigned* mmax   = (unsigned*)alloc(4);
    unsigned* stbits = (unsigned*)alloc(3 * 4);
    float*    posA   = (float*)alloc((size_t)1372 * 3 * 4);
    float*    posB   = (float*)alloc((size_t)1372 * 3 * 4);
    float*    nmA    = (float*)alloc(1372 * 4);
    float*    nmB    = (float*)alloc(1372 * 4);
    int*      clbuf  = (int*)alloc((size_t)NPTS * 4);
    float*    pcnt   = (float*)alloc(1372 * 4);
    float*    psum   = (float*)alloc((size_t)1372 * 3 * 4);
    unsigned* nxenc  = (unsigned*)alloc((size_t)256 * 512 * 4);   // also covers 1372*64
    unsigned* bitmap = (unsigned*)alloc(((size_t)1372 * 1372 / 32 + 2) * 4);
    int*      sA     = (int*)alloc((size_t)EDGES * 4);
    int*      dA     = (int*)alloc((size_t)EDGES * 4);
    int*      sB     = (int*)alloc((size_t)EDGES * 4);
    int*      dB     = (int*)alloc((size_t)EDGES * 4);
    float*    emA    = (float*)alloc((size_t)EDGES * 4);
    float*    emB    = (float*)alloc((size_t)EDGES * 4);
    float*    px     = (float*)alloc((size_t)256 * 512 * 4);
    float*    h1c    = (float*)alloc((size_t)BATCH * 1024 * 4);
    float*    h1     = (float*)alloc((size_t)BATCH * 1024 * 4);
    float*    l2c    = (float*)alloc((size_t)BATCH * NCLS * 4);
    (void)ws_size; (void)n_in; (void)in_sizes; (void)out_size;

    const float* xp   = x0;
    const float* posp = pos0;
    const float* nmp  = nullptr;   // level-0 masks are all ones
    const int*   sp   = src0;
    const int*   dp   = dst0;
    const float* emp  = nullptr;

    int cpg_prev = NPG;            // batch divisor of current node level

    for (int l = 0; l < 4; ++l) {
        const int n = nlist[l], cin = cinl[l], cout = coutl[l];
        const long long nKc = (long long)n * KD * cin;

        // ---- edge pseudo-coordinates ----
        fill_u32<<<1, 1, 0, stream>>>(mmax, 0u, 1);
        edge_absmax<<<gdiv(EDGES, 256), 256, 0, stream>>>(posp, sp, dp, emp, mmax, EDGES);
        edge_attr  <<<gdiv(EDGES, 256), 256, 0, stream>>>(posp, sp, dp, mmax, attr, EDGES);

        // ---- spline scatter into A + degree ----
        fill_f32<<<gdiv(nKc, 256), 256, 0, stream>>>(Abuf, 0.f, nKc);
        fill_f32<<<gdiv(n, 256),  256, 0, stream>>>(deg, 0.f, n);
        spline_scatter<<<gdiv((long long)EDGES * cin, 256), 256, 0, stream>>>(
            xp, attr, sp, dp, emp, Abuf, deg, EDGES, cin);

        // ---- WMMA GEMM: conv = A(n, KD*cin) @ W(KD*cin, cout), split-K for skinny M ----
        const int Kdim = KD * cin;
        const int zsplit = (l == 0) ? 1 : 8;
        int kchunk = (((Kdim + zsplit - 1) / zsplit) + (TK - 1)) & ~(TK - 1);
        if (zsplit > 1)
            fill_f32<<<gdiv((long long)n * cout, 256), 256, 0, stream>>>(conv, 0.f, (long long)n * cout);
        dim3 gg(gdiv(cout, TN), gdiv(n, TM), zsplit);
        wmma_gemm_f32<<<gg, 256, 0, stream>>>(Abuf, W[l], conv, n, Kdim, cout, kchunk);

        // ---- epilogue + BN ----
        post_conv<<<gdiv((long long)n * cout, 256), 256, 0, stream>>>(
            conv, xp, R[l], Bi[l], deg, tbuf, n, cin, cout);
        bn_stats<<<cout, 256, 0, stream>>>(tbuf, nmp, bnsum, bnsq, bncnt, n, cout);
        bn_apply<<<gdiv((long long)n * cout, 256), 256, 0, stream>>>(
            tbuf, nmp, bnsum, bnsq, bncnt, Gm[l], Be[l], xA, n, cout);

        if (l < 3) {
            const int G = GRD[l], cpg = cpgl[l], Nn = Nnl[l];
            float* posN = (l & 1) ? posB : posA;         // l=0 -> posA, l=1 -> posB, l=2 -> posA
            float* nmN  = (l & 1) ? nmB  : nmA;
            int*   sN   = (l & 1) ? sB : sA;
            int*   dN   = (l & 1) ? dB : dA;
            float* emN  = (l & 1) ? emB : emA;

            fill_u32<<<1, 1, 0, stream>>>(stbits, 0x7f800000u, 3);
            pos_min<<<gdiv(n, 256), 256, 0, stream>>>(posp, nmp, stbits, n);
            fill_f32<<<gdiv(Nn, 256), 256, 0, stream>>>(pcnt, 0.f, Nn);
            fill_f32<<<gdiv((long long)Nn * 3, 256), 256, 0, stream>>>(psum, 0.f, (long long)Nn * 3);
            fill_u32<<<gdiv((long long)Nn * cout, 256), 256, 0, stream>>>(
                nxenc, ENC_NEG_FLT_MAX, (long long)Nn * cout);
            pool_node<<<gdiv(n, 256), 256, 0, stream>>>(
                posp, nmp, stbits, clbuf, pcnt, psum, n, cpg_prev, cpg, G, VOX[l]);
            pool_xmax<<<gdiv((long long)n * cout, 256), 256, 0, stream>>>(xA, nmp, clbuf, nxenc, n, cout);
            pool_fin_x<<<gdiv((long long)Nn * cout, 256), 256, 0, stream>>>(nxenc, pcnt, xB, Nn, cout);
            pool_fin_node<<<gdiv(Nn, 256), 256, 0, stream>>>(pcnt, psum, posN, nmN, Nn);

            long long words = ((long long)Nn * Nn + 31) / 32;
            fill_u32<<<gdiv(words, 256), 256, 0, stream>>>(bitmap, 0u, words);
            edge_remap<<<gdiv(EDGES, 256), 256, 0, stream>>>(
                sp, dp, emp, clbuf, sN, dN, emN, bitmap, EDGES, Nn);

            xp = xB; posp = posN; nmp = nmN; sp = sN; dp = dN; emp = emN;
            cpg_prev = cpg;
        }
    }

    // ---- head pooling: G=2, voxel 80, cl = batch*SLOTS + vid  (256 clusters x 512 ch) ----
    {
        const int n = 108, cout = 512, Nn = BATCH * SLOTS;
        fill_u32<<<1, 1, 0, stream>>>(stbits, 0x7f800000u, 3);
        pos_min<<<gdiv(n, 256), 256, 0, stream>>>(posp, nmp, stbits, n);
        fill_f32<<<gdiv(Nn, 256), 256, 0, stream>>>(pcnt, 0.f, Nn);
        fill_f32<<<gdiv((long long)Nn * 3, 256), 256, 0, stream>>>(psum, 0.f, (long long)Nn * 3);
        fill_u32<<<gdiv((long long)Nn * cout, 256), 256, 0, stream>>>(
            nxenc, ENC_NEG_FLT_MAX, (long long)Nn * cout);
        pool_node<<<gdiv(n, 256), 256, 0, stream>>>(
            posp, nmp, stbits, clbuf, pcnt, psum, n, cpg_prev, SLOTS, 2, 80.f);
        pool_xmax<<<gdiv((long long)n * cout, 256), 256, 0, stream>>>(xA, nmp, clbuf, nxenc, n, cout);
        pool_fin_x<<<gdiv((long long)Nn * cout, 256), 256, 0, stream>>>(nxenc, pcnt, px, Nn, cout);
    }

    // ---- FC1: (4 x 32768) @ fw1 (32768 x 1024), split-K 32 ----
    fill_f32<<<gdiv(BATCH * 1024, 256), 256, 0, stream>>>(h1c, 0.f, BATCH * 1024);
    wmma_gemm_f32<<<dim3(1024 / TN, 1, 32), 256, 0, stream>>>(px, fw1, h1c, BATCH, SLOTS * 512, 1024, 1024);
    fc_elu<<<gdiv(BATCH * 1024, 256), 256, 0, stream>>>(h1c, fb1, h1, BATCH, 1024);

    // ---- FC2: (4 x 1024) @ fw2 (1024 x 101) ----
    fill_f32<<<gdiv(BATCH * NCLS, 256), 256, 0, stream>>>(l2c, 0.f, BATCH * NCLS);
    wmma_gemm_f32<<<dim3(2, 1, 1), 256, 0, stream>>>(h1, fw2, l2c, BATCH, 1024, NCLS, 1024);

    head_logsoftmax<<<1, 128, 0, stream>>>(l2c, fb2, (float*)d_out);
}